// QuantumCircuitGNN_22711787061446
// MI455X (gfx1250) — compile-verified
//
#include <hip/hip_runtime.h>
#include <math.h>

typedef __attribute__((ext_vector_type(2))) float v2f;
typedef __attribute__((ext_vector_type(8))) float v8f;

#define HDIM 64
#define GRAPHS 128
#define EPS 1e-5f

// ---------------- degree / norm ----------------
__global__ void deg_init_kernel(float* deg, int n) {
  int i = blockIdx.x * blockDim.x + threadIdx.x;
  if (i < n) deg[i] = 1.0f;  // self-loop
}

__global__ void deg_scatter_kernel(const int* __restrict__ dst, float* deg, int E) {
  int e = blockIdx.x * blockDim.x + threadIdx.x;
  if (e < E) atomicAdd(&deg[dst[e]], 1.0f);
}

__global__ void deg_finalize_kernel(float* deg, int n) {
  int i = blockIdx.x * blockDim.x + threadIdx.x;
  if (i < n) deg[i] = rsqrtf(deg[i]);  // deg >= 1 always
}

// ---------------- dense GEMM: out[N x 64] = A[N x K] @ W[K x 64] (f32 WMMA) ----------------
// blockDim = 128 (4 waves). Each wave computes a 16-row x 64-col tile with
// four V_WMMA_F32_16X16X4_F32 accumulators, K-stepped by 4.
// W is staged in LDS pair-interleaved: k-pair p at sW[p*128 + 2*col + (k&1)],
// so every B fragment is one contiguous, 8B-aligned ds_load_b64.
__global__ void gcn_gemm_wmma_kernel(const float* __restrict__ A,
                                     const float* __restrict__ W,
                                     float* __restrict__ out, int n, int K) {
  extern __shared__ float sW[];  // (K/2) * 128 floats
  for (int idx = threadIdx.x; idx < K * HDIM; idx += blockDim.x) {
    const int k = idx >> 6;        // row of W
    const int c = idx & 63;        // col of W
    sW[(k >> 1) * 128 + c * 2 + (k & 1)] = W[idx];
  }
  __syncthreads();

  const int wave = threadIdx.x >> 5;
  const int lane = threadIdx.x & 31;
  const int rowBase = (blockIdx.x * 4 + wave) * 16;
  const int m = lane & 15;   // M for A-frag, N-col for B/C-frag
  const int kh = lane >> 4;  // K-half select (0: K=0,1 ; 1: K=2,3)
  const int row = rowBase + m;
  const bool valid = row < n;
  const float mask = valid ? 1.0f : 0.0f;
  const long long arow = (long long)(valid ? row : (n - 1)) * K;  // clamped: always in-bounds

  v8f acc0 = {}, acc1 = {}, acc2 = {}, acc3 = {};
  for (int k = 0; k < K; k += 4) {
    const int ka = k + kh * 2;  // even -> 8B-aligned
    v2f a = *reinterpret_cast<const v2f*>(A + arow + ka);
    a = a * mask;  // zero out-of-range rows without exec divergence
    const float* wp = &sW[(ka >> 1) * 128];
    const v2f b0 = *reinterpret_cast<const v2f*>(wp + 2 * m);
    const v2f b1 = *reinterpret_cast<const v2f*>(wp + 2 * (16 + m));
    const v2f b2 = *reinterpret_cast<const v2f*>(wp + 2 * (32 + m));
    const v2f b3 = *reinterpret_cast<const v2f*>(wp + 2 * (48 + m));
    acc0 = __builtin_amdgcn_wmma_f32_16x16x4_f32(false, a, false, b0, (short)0, acc0, false, false);
    acc1 = __builtin_amdgcn_wmma_f32_16x16x4_f32(false, a, false, b1, (short)0, acc1, false, false);
    acc2 = __builtin_amdgcn_wmma_f32_16x16x4_f32(false, a, false, b2, (short)0, acc2, false, false);
    acc3 = __builtin_amdgcn_wmma_f32_16x16x4_f32(false, a, false, b3, (short)0, acc3, false, false);
  }

  // C/D layout: VGPR r holds M = r (lanes 0-15) / r+8 (lanes 16-31), N = lane&15
#pragma unroll
  for (int r = 0; r < 8; ++r) {
    const int ro = rowBase + r + 8 * kh;
    if (ro < n) {
      const long long o = (long long)ro * HDIM + m;
      out[o]      = acc0[r];
      out[o + 16] = acc1[r];
      out[o + 32] = acc2[r];
      out[o + 48] = acc3[r];
    }
  }
}

// ---------------- aggregation ----------------
__global__ void agg_init_kernel(const float* __restrict__ XW, const float* __restrict__ dis,
                                float* __restrict__ agg, int n) {
  int t = blockIdx.x * blockDim.x + threadIdx.x;
  if (t < n * HDIM) {
    int i = t >> 6;
    float d = dis[i];
    agg[t] = d * d * XW[t];  // self-loop term
  }
}

__global__ void agg_scatter_kernel(const float* __restrict__ XW, const float* __restrict__ dis,
                                   const int* __restrict__ src, const int* __restrict__ dst,
                                   float* agg, int E) {
  int t = blockIdx.x * blockDim.x + threadIdx.x;
  int e = t >> 4;
  if (e >= E) return;
  int f = (t & 15) * 4;
  int s = src[e], d = dst[e];
  float w = dis[s] * dis[d];
  const float4 v = *reinterpret_cast<const float4*>(XW + (long long)s * HDIM + f);
  float* p = agg + (long long)d * HDIM + f;
  atomicAdd(p + 0, w * v.x);
  atomicAdd(p + 1, w * v.y);
  atomicAdd(p + 2, w * v.z);
  atomicAdd(p + 3, w * v.w);
}

__global__ void bn_relu_kernel(float* agg, const float* __restrict__ b,
                               const float* __restrict__ g, const float* __restrict__ be,
                               const float* __restrict__ rm, const float* __restrict__ rv, int n) {
  int t = blockIdx.x * blockDim.x + threadIdx.x;
  if (t < n * HDIM) {
    int j = t & 63;
    float v = agg[t] + b[j];
    v = (v - rm[j]) * rsqrtf(rv[j] + EPS) * g[j] + be[j];
    agg[t] = fmaxf(v, 0.0f);
  }
}

// ---------------- pooling ----------------
__global__ void pool_clear_kernel(float* sums, float* cnts) {
  int t = blockIdx.x * blockDim.x + threadIdx.x;
  if (t < GRAPHS * HDIM) sums[t] = 0.0f;
  if (t < GRAPHS) cnts[t] = 0.0f;
}

__global__ void pool_scatter_kernel(const float* __restrict__ h, const int* __restrict__ batch,
                                    float* sums, float* cnts, int n) {
  int t = blockIdx.x * blockDim.x + threadIdx.x;
  if (t < n * HDIM) {
    int i = t >> 6, j = t & 63;
    int g = batch[i];
    atomicAdd(&sums[g * HDIM + j], h[t]);
    if (j == 0) atomicAdd(&cnts[g], 1.0f);
  }
}

// ---------------- heads ----------------
__global__ void heads_kernel(const float* __restrict__ sums, const float* __restrict__ cnts,
                             const float* __restrict__ sw1, const float* __restrict__ sb1,
                             const float* __restrict__ sw2, const float* __restrict__ sb2,
                             const float* __restrict__ aw1, const float* __restrict__ ab1,
                             const float* __restrict__ aw2, const float* __restrict__ ab2,
                             float* __restrict__ out) {
  __shared__ float s_emb[HDIM];
  __shared__ float s_h1[32];
  __shared__ float s_h2[32];
  int g = blockIdx.x, t = threadIdx.x;
  float cnt = fmaxf(cnts[g], 1.0f);
  s_emb[t] = sums[g * HDIM + t] / cnt;
  __syncthreads();
  if (t < 32) {
    float a1 = sb1[t], a2 = ab1[t];
    for (int j = 0; j < HDIM; ++j) {
      float e = s_emb[j];
      a1 += e * sw1[j * 32 + t];
      a2 += e * aw1[j * 32 + t];
    }
    s_h1[t] = fmaxf(a1, 0.0f);
    s_h2[t] = fmaxf(a2, 0.0f);
  }
  __syncthreads();
  if (t == 0) {
    float sc = sb2[0];
    for (int k = 0; k < 32; ++k) sc += s_h1[k] * sw2[k];
    out[g] = 1.0f / (1.0f + __expf(-sc));  // score: out[0..G)
  }
  if (t < 4) {
    float ac = ab2[t];
    for (int k = 0; k < 32; ++k) ac += s_h2[k] * aw2[k * 4 + t];
    out[GRAPHS + g * 4 + t] = ac;  // actions: out[G .. G+4G)
  }
}

extern "C" void kernel_launch(void* const* d_in, const int* in_sizes, int n_in,
                              void* d_out, int out_size, void* d_ws, size_t ws_size,
                              hipStream_t stream) {
  const float* x    = (const float*)d_in[0];
  const int* ei     = (const int*)d_in[1];
  const int* batch  = (const int*)d_in[2];
  const int N = in_sizes[0] / 20;
  const int E = in_sizes[1] / 2;
  const int* src = ei;
  const int* dst = ei + E;

  float* ws   = (float*)d_ws;
  float* dis  = ws;                                 // N
  float* buf1 = dis + N;                            // N*64  (XW)
  float* buf2 = buf1 + (size_t)N * HDIM;            // N*64  (agg -> h, in place)
  float* sums = buf2 + (size_t)N * HDIM;            // G*64
  float* cnts = sums + GRAPHS * HDIM;               // G

  const int NT = 256;
  // degrees -> dis = rsqrt(deg)
  deg_init_kernel<<<(N + NT - 1) / NT, NT, 0, stream>>>(dis, N);
  deg_scatter_kernel<<<(E + NT - 1) / NT, NT, 0, stream>>>(dst, dis, E);
  deg_finalize_kernel<<<(N + NT - 1) / NT, NT, 0, stream>>>(dis, N);

  const float* hin = x;
  int K = 20;
  const int nh = N * HDIM;
  for (int layer = 0; layer < 3; ++layer) {
    const float* Wl  = (const float*)d_in[4 + layer * 2];
    const float* bl  = (const float*)d_in[5 + layer * 2];
    const float* gl  = (const float*)d_in[10 + layer * 4];
    const float* bel = (const float*)d_in[11 + layer * 4];
    const float* rml = (const float*)d_in[12 + layer * 4];
    const float* rvl = (const float*)d_in[13 + layer * 4];

    gcn_gemm_wmma_kernel<<<(N + 63) / 64, 128, K * HDIM * sizeof(float), stream>>>(
        hin, Wl, buf1, N, K);
    agg_init_kernel<<<(nh + NT - 1) / NT, NT, 0, stream>>>(buf1, dis, buf2, N);
    agg_scatter_kernel<<<((E * 16) + NT - 1) / NT, NT, 0, stream>>>(buf1, dis, src, dst, buf2, E);
    bn_relu_kernel<<<(nh + NT - 1) / NT, NT, 0, stream>>>(buf2, bl, gl, bel, rml, rvl, N);
    hin = buf2;
    K = HDIM;
  }

  pool_clear_kernel<<<(GRAPHS * HDIM + GRAPHS + NT - 1) / NT, NT, 0, stream>>>(sums, cnts);
  pool_scatter_kernel<<<(nh + NT - 1) / NT, NT, 0, stream>>>(buf2, batch, sums, cnts, N);

  heads_kernel<<<GRAPHS, HDIM, 0, stream>>>(
      sums, cnts,
      (const float*)d_in[22], (const float*)d_in[23],
      (const float*)d_in[24], (const float*)d_in[25],
      (const float*)d_in[26], (const float*)d_in[27],
      (const float*)d_in[28], (const float*)d_in[29],
      (float*)d_out);
}